// TemporalSpikingAttention_9234179687029
// MI455X (gfx1250) — compile-verified
//
#include <hip/hip_runtime.h>
#include <hip/hip_bf16.h>

typedef __attribute__((ext_vector_type(2))) float v2f;
typedef __attribute__((ext_vector_type(8))) float v8f;

#define BB 2
#define HH 16
#define SS 2048
#define DD 64
#define TM 16                 // query rows per workgroup tile
#define THREADS 256
#define WAVES 8
#define NK_PER_WAVE (SS / WAVES)   // 256 keys per wave
#define SROW (SS + 4)              // padded LDS row stride (floats) -> conflict-free
#define PSTRIDE 68                 // padded partial row stride
#define SCALE 0.03125f             // 1/sqrt(1024) == 2^-5 exact

__global__ __launch_bounds__(THREADS)
void tsa_fused_kernel(const float* __restrict__ Q,
                      const float* __restrict__ K,
                      const float* __restrict__ V,
                      const float* __restrict__ Noise,
                      const float* __restrict__ tdecay_p,
                      float* __restrict__ out_spikes,
                      float* __restrict__ out_attn)
{
    __shared__ __align__(16) float lds_s[TM * SROW];   // scores/w tile, reused for AV partials
    __shared__ float lds_decay[SS];
    __shared__ float lds_red[TM * 17];
    __shared__ float lds_rdenom[TM];                   // per-row 1/(Sw + 1e-8)

    const int tid  = threadIdx.x;
    const int wave = tid >> 5;
    const int lane = tid & 31;
    const int lrow = lane & 15;   // M (or N) index inside 16-wide fragment
    const int hi   = lane >> 4;   // selects K pair {2*hi, 2*hi+1} in fp32 fragments

    const int q0 = blockIdx.x * TM;        // 0..2047 step 16
    const int bh = blockIdx.y;             // 0..31
    const size_t base = (size_t)bh * SS * DD;   // row-major [S][D] per (b,h)

    // ---- decay table: 0.95^(S-t) ----
    const float td = tdecay_p[0];
    for (int t = tid; t < SS; t += THREADS)
        lds_decay[t] = powf(td, (float)(SS - t));

    // =====================  QK^T via v_wmma_f32_16x16x4_f32  =====================
    v2f qfrag[16];
    {
        const float* qrow = Q + base + (size_t)(q0 + lrow) * DD + 2 * hi;
        #pragma unroll
        for (int kk = 0; kk < 16; ++kk)
            qfrag[kk] = *(const v2f*)(qrow + 4 * kk);
    }

    const int nbase = wave * NK_PER_WAVE;
    for (int nt = 0; nt < NK_PER_WAVE / 16; ++nt) {
        const int n0 = nbase + nt * 16;
        const float* krow = K + base + (size_t)(n0 + lrow) * DD + 2 * hi;
        v8f acc = {0.f,0.f,0.f,0.f,0.f,0.f,0.f,0.f};
        #pragma unroll
        for (int kk = 0; kk < 16; ++kk) {
            v2f bfrag = *(const v2f*)(krow + 4 * kk);
            acc = __builtin_amdgcn_wmma_f32_16x16x4_f32(
                false, qfrag[kk], false, bfrag, (short)0, acc, false, false);
        }
        // D layout: reg r holds row (r + 8*hi), col n0+lrow
        #pragma unroll
        for (int r = 0; r < 8; ++r)
            lds_s[(r + 8 * hi) * SROW + n0 + lrow] = acc[r] * SCALE;
    }
    __syncthreads();

    // =====================  softmax + decay (in LDS, per-row scalars)  =====================
    const int srow = tid >> 4;    // 16 threads cooperate per row
    const int scol = tid & 15;
    float* rowp = lds_s + srow * SROW;

    // pass A: row max
    float m = -3.402823466e38f;
    for (int j = scol; j < SS; j += 16) m = fmaxf(m, rowp[j]);
    lds_red[srow * 17 + scol] = m;
    __syncthreads();
    float rmax = lds_red[srow * 17];
    #pragma unroll
    for (int i = 1; i < 16; ++i) rmax = fmaxf(rmax, lds_red[srow * 17 + i]);
    __syncthreads();

    // pass B: exp and sum
    float s = 0.f;
    for (int j = scol; j < SS; j += 16) {
        float p = __expf(rowp[j] - rmax);
        rowp[j] = p;
        s += p;
    }
    lds_red[srow * 17 + scol] = s;
    __syncthreads();
    float Sp = 0.f;
    #pragma unroll
    for (int i = 0; i < 16; ++i) Sp += lds_red[srow * 17 + i];
    __syncthreads();

    // pass C: w = (p/Sp)*decay via reciprocal multiply; accumulate row sum of w
    const float invSp = 1.0f / Sp;
    float sw = 0.f;
    for (int j = scol; j < SS; j += 16) {
        float w = (rowp[j] * invSp) * lds_decay[j];
        rowp[j] = w;
        sw += w;
    }
    lds_red[srow * 17 + scol] = sw;
    __syncthreads();
    float Sw = 0.f;
    #pragma unroll
    for (int i = 0; i < 16; ++i) Sw += lds_red[srow * 17 + i];
    if (scol == 0) lds_rdenom[srow] = 1.0f / (Sw + 1e-8f);
    __syncthreads();

    // =====================  coalesced attn writeback (float4, denom folded in)  ============
    {
        float* attn_out = out_attn + ((size_t)bh * SS + q0) * SS;
        for (int idx = tid; idx < TM * (SS / 4); idx += THREADS) {
            int row = idx >> 9;          // / 512
            int c4  = idx & 511;
            float rd = lds_rdenom[row];
            float4 v = *(const float4*)(lds_s + row * SROW + c4 * 4);
            v.x *= rd; v.y *= rd; v.z *= rd; v.w *= rd;
            *(float4*)(attn_out + (size_t)row * SS + c4 * 4) = v;
        }
    }

    // =====================  AV = w @ V via WMMA (per-wave K-split); /denom later ===========
    v8f avacc[4];
    #pragma unroll
    for (int n = 0; n < 4; ++n) avacc[n] = (v8f){0.f,0.f,0.f,0.f,0.f,0.f,0.f,0.f};

    const int kb0 = wave * NK_PER_WAVE;
    const float* vbase = V + base;
    for (int kk = 0; kk < NK_PER_WAVE / 4; ++kk) {
        const int k = kb0 + 4 * kk + 2 * hi;
        v2f a = *(const v2f*)(lds_s + lrow * SROW + k);          // w A-fragment
        const float* vr = vbase + (size_t)k * DD + lrow;
        #pragma unroll
        for (int n = 0; n < 4; ++n) {
            v2f bfrag;
            bfrag.x = vr[n * 16];        // V[k][n*16+lrow]
            bfrag.y = vr[DD + n * 16];   // V[k+1][n*16+lrow]
            avacc[n] = __builtin_amdgcn_wmma_f32_16x16x4_f32(
                false, a, false, bfrag, (short)0, avacc[n], false, false);
        }
    }
    __syncthreads();   // all LDS reads of w complete -> safe to overwrite with partials

    // store per-wave partials (padded stride, conflict-free)
    {
        float* part = lds_s + wave * (TM * PSTRIDE);
        #pragma unroll
        for (int n = 0; n < 4; ++n)
            #pragma unroll
            for (int r = 0; r < 8; ++r)
                part[(r + 8 * hi) * PSTRIDE + n * 16 + lrow] = avacc[n][r];
    }
    __syncthreads();

    // =====================  cross-wave reduce + spike epilogue  =====================
    for (int idx = tid; idx < TM * DD; idx += THREADS) {
        int row = idx >> 6;
        int col = idx & 63;
        float av = 0.f;
        #pragma unroll
        for (int w = 0; w < WAVES; ++w)
            av += lds_s[w * (TM * PSTRIDE) + row * PSTRIDE + col];
        av *= lds_rdenom[row];           // apply per-row renormalization

        float prob = 1.0f / (1.0f + __expf(-av));
        size_t gi = ((size_t)bh * SS + q0 + row) * DD + col;
        float bern = (Noise[gi] < prob) ? 1.0f : 0.0f;
        // replicate clip(bern + sg - sg, 0, 1) rounding exactly
        float sg = fminf(fmaxf(1.0f - fabsf(av), 0.0f), 1.0f) * 0.1f;
        float spike = bern + sg;
        spike = spike - sg;
        spike = fminf(fmaxf(spike, 0.0f), 1.0f);
        out_spikes[gi] = spike;
    }
}

extern "C" void kernel_launch(void* const* d_in, const int* in_sizes, int n_in,
                              void* d_out, int out_size, void* d_ws, size_t ws_size,
                              hipStream_t stream) {
    (void)in_sizes; (void)n_in; (void)out_size; (void)d_ws; (void)ws_size;
    const float* Q  = (const float*)d_in[0];
    const float* K  = (const float*)d_in[1];
    const float* V  = (const float*)d_in[2];
    const float* Nz = (const float*)d_in[3];
    const float* td = (const float*)d_in[4];

    float* out_spikes = (float*)d_out;                                  // [B,H,S,D]
    float* out_attn   = out_spikes + (size_t)BB * HH * SS * DD;         // [B,H,S,S]

    dim3 grid(SS / TM, BB * HH);   // 128 x 32
    tsa_fused_kernel<<<grid, THREADS, 0, stream>>>(Q, K, V, Nz, td, out_spikes, out_attn);
}